// S2M2CinematicLoss_13804024889847
// MI455X (gfx1250) — compile-verified
//
#include <hip/hip_runtime.h>
#include <hip/hip_bf16.h>
#include <math.h>

typedef __attribute__((ext_vector_type(2))) float v2f;
typedef __attribute__((ext_vector_type(8))) float v8f;

constexpr int B  = 4;
constexpr int H  = 544;
constexpr int W  = 960;
constexpr int H2 = 272;   // half-res
constexpr int W2 = 480;

constexpr float ALPHA   = 0.85f;
constexpr float GT_TH   = 2.0f;
constexpr float CONF_TH = 0.3f;
constexpr float SIGN_W  = 0.3f, MAG_W = 0.7f;
constexpr float W_GT = 1.0f, W_PHOTO = 1.0f, W_SIGNMAG = 0.5f, W_SMOOTH = 0.1f;
constexpr float SSIM_C1 = 0.0001f, SSIM_C2 = 0.0009f;

// ws float layout: [0]=gt_num [1]=gt_den [2]=ph_num [3]=ph_den
//                  [4]=smooth_dx_sum [5]=smooth_dy_sum
//                  [6]=sign_sum [7]=mag_sum [8]=active_cnt
// [16 ..)            lg (B*H2*W2), then rg (B*H2*W2)

// ---------------------------------------------------------------- init
__global__ void init_kernel(float* acc) {
  if (threadIdx.x < 16) acc[threadIdx.x] = 0.0f;
}

// ---------------------------------------------------------------- block reduce helper
__device__ inline void block_reduce_atomic(float v, float* red, float* target) {
  const int tid = threadIdx.x;
  red[tid] = v;
  __syncthreads();
  for (int s = blockDim.x >> 1; s > 0; s >>= 1) {
    if (tid < s) red[tid] += red[tid + s];
    __syncthreads();
  }
  if (tid == 0) atomicAdd(target, red[0]);
  __syncthreads();
}

// ---------------------------------------------------------------- gt anchor + smoothness
__global__ __launch_bounds__(256) void point_kernel(const float* __restrict__ pred,
                                                    const float* __restrict__ gt,
                                                    const float* __restrict__ conf,
                                                    const float* __restrict__ occ,
                                                    const float* __restrict__ left,
                                                    float* acc) {
  __shared__ float red[256];
  const int idx = blockIdx.x * 256 + threadIdx.x;
  float gnum = 0.f, gden = 0.f, sx = 0.f, sy = 0.f;
  if (idx < B * H * W) {
    const int x = idx % W;
    const int y = (idx / W) % H;
    const int b = idx / (H * W);
    const float p = pred[idx];
    const float g = gt[idx];
    const float pos = (g > GT_TH) ? 1.f : 0.f;
    const float trust = conf[idx] * occ[idx] * pos;
    gnum = trust * fabsf(p - g);
    gden = trust;
    const float* lb = left + (size_t)b * 3 * H * W;
    if (x < W - 1) {
      const float ddx = fabsf(pred[idx + 1] - p);
      float gi = 0.f;
#pragma unroll
      for (int c = 0; c < 3; ++c) {
        const size_t o = ((size_t)c * H + y) * W + x;
        gi += fabsf(lb[o + 1] - lb[o]);
      }
      sx = ddx * __expf(-gi * (1.f / 3.f));
    }
    if (y < H - 1) {
      const float ddy = fabsf(pred[idx + W] - p);
      float gi = 0.f;
#pragma unroll
      for (int c = 0; c < 3; ++c) {
        const size_t o = ((size_t)c * H + y) * W + x;
        gi += fabsf(lb[o + W] - lb[o]);
      }
      sy = ddy * __expf(-gi * (1.f / 3.f));
    }
  }
  block_reduce_atomic(gnum, red, &acc[0]);
  block_reduce_atomic(gden, red, &acc[1]);
  block_reduce_atomic(sx,   red, &acc[4]);
  block_reduce_atomic(sy,   red, &acc[5]);
}

// ---------------------------------------------------------------- photometric (warp + SSIM)
__global__ __launch_bounds__(256) void photo_kernel(const float* __restrict__ pred,
                                                    const float* __restrict__ left,
                                                    const float* __restrict__ right,
                                                    float* acc) {
  __shared__ float sWp[3][18][18];
  __shared__ float sLf[3][18][18];
  __shared__ float sVd[18][18];
  __shared__ float red[256];
  const int b   = blockIdx.z;
  const int by0 = blockIdx.y * 16;
  const int bx0 = blockIdx.x * 16;
  const int tid = threadIdx.x;
  const float* pb = pred  + (size_t)b * H * W;
  const float* lb = left  + (size_t)b * 3 * H * W;
  const float* rb = right + (size_t)b * 3 * H * W;

  for (int idx = tid; idx < 18 * 18; idx += 256) {
    const int py = idx / 18, px = idx % 18;
    const int gy = by0 - 1 + py, gx = bx0 - 1 + px;
    float w0 = 0.f, w1 = 0.f, w2 = 0.f, l0 = 0.f, l1v = 0.f, l2 = 0.f, vd = 0.f;
    if (gy >= 0 && gy < H && gx >= 0 && gx < W) {
      const float dsp = pb[gy * W + gx];
      const float xs  = (float)gx - dsp;
      vd = (xs > 0.f && xs < (float)(W - 1)) ? 1.f : 0.f;
      const float xc  = fminf(fmaxf(xs, 0.f), (float)(W - 1));
      const float x0f = floorf(xc);
      const float wf  = xc - x0f;
      const int x0 = (int)x0f;
      const int x1 = min(x0 + 1, W - 1);
      const size_t row0 = (size_t)(0 * H + gy) * W;
      const size_t row1 = (size_t)(1 * H + gy) * W;
      const size_t row2 = (size_t)(2 * H + gy) * W;
      w0 = rb[row0 + x0] * (1.f - wf) + rb[row0 + x1] * wf;
      w1 = rb[row1 + x0] * (1.f - wf) + rb[row1 + x1] * wf;
      w2 = rb[row2 + x0] * (1.f - wf) + rb[row2 + x1] * wf;
      l0 = lb[row0 + gx]; l1v = lb[row1 + gx]; l2 = lb[row2 + gx];
    }
    sWp[0][py][px] = w0; sWp[1][py][px] = w1; sWp[2][py][px] = w2;
    sLf[0][py][px] = l0; sLf[1][py][px] = l1v; sLf[2][py][px] = l2;
    sVd[py][px] = vd;
  }
  __syncthreads();

  const int py = 1 + tid / 16, px = 1 + tid % 16;
  float ss = 0.f, l1sum = 0.f;
#pragma unroll
  for (int c = 0; c < 3; ++c) {
    float sxs = 0.f, sys = 0.f, sxx = 0.f, syy = 0.f, sxy = 0.f;
#pragma unroll
    for (int dy = -1; dy <= 1; ++dy)
#pragma unroll
      for (int dx = -1; dx <= 1; ++dx) {
        const float xv = sLf[c][py + dy][px + dx];
        const float yv = sWp[c][py + dy][px + dx];
        sxs += xv; sys += yv;
        sxx += xv * xv; syy += yv * yv; sxy += xv * yv;
      }
    const float mx = sxs * (1.f / 9.f), my = sys * (1.f / 9.f);
    const float vx = fmaxf(sxx * (1.f / 9.f) - mx * mx, 0.f);
    const float vy = fmaxf(syy * (1.f / 9.f) - my * my, 0.f);
    const float cxy = sxy * (1.f / 9.f) - mx * my;
    const float nn = (2.f * mx * my + SSIM_C1) * (2.f * cxy + SSIM_C2);
    const float dd = (mx * mx + my * my + SSIM_C1) * (vx + vy + SSIM_C2);
    ss += fminf(fmaxf((1.f - nn / dd) * 0.5f, 0.f), 1.f);
    l1sum += fabsf(sLf[c][py][px] - sWp[c][py][px]);
  }
  ss *= (1.f / 3.f);
  l1sum *= (1.f / 3.f);
  const float err = ALPHA * ss + (1.f - ALPHA) * l1sum;
  const float vd = sVd[py][px];
  block_reduce_atomic(err * vd, red, &acc[2]);
  block_reduce_atomic(vd,       red, &acc[3]);
}

// ---------------------------------------------------------------- half-res grayscale
__global__ __launch_bounds__(256) void downsample_kernel(const float* __restrict__ left,
                                                         const float* __restrict__ right,
                                                         float* __restrict__ lg,
                                                         float* __restrict__ rg) {
  const int idx = blockIdx.x * 256 + threadIdx.x;
  if (idx >= B * H2 * W2) return;
  const int x = idx % W2;
  const int y = (idx / W2) % H2;
  const int b = idx / (W2 * H2);
  float sl = 0.f, sr = 0.f;
#pragma unroll
  for (int c = 0; c < 3; ++c)
#pragma unroll
    for (int dy = 0; dy < 2; ++dy)
#pragma unroll
      for (int dx = 0; dx < 2; ++dx) {
        const size_t o = (((size_t)b * 3 + c) * H + 2 * y + dy) * W + 2 * x + dx;
        sl += left[o];
        sr += right[o];
      }
  lg[idx] = sl * (1.f / 12.f);
  rg[idx] = sr * (1.f / 12.f);
}

// ---------------------------------------------------------------- WMMA helper
__device__ inline v8f wmma_f32(v2f a, v2f b, v8f c) {
  return __builtin_amdgcn_wmma_f32_16x16x4_f32(false, a, false, b, (short)0, c, false, false);
}

// ---------------------------------------------------------------- NCC via WMMA box filters + sign/mag loss
__global__ __launch_bounds__(128) void ncc_signmag_kernel(const float* __restrict__ pred,
                                                          const float* __restrict__ lg,
                                                          const float* __restrict__ rg,
                                                          float* acc) {
  const int b  = blockIdx.z;
  const int oy = blockIdx.y * 16;   // half-res tile origin
  const int ox = blockIdx.x * 16;
  const int tid  = threadIdx.x;
  const int lane = tid & 31;
  const int wv   = tid >> 5;        // 4 waves
  const int hi   = lane >> 4;       // lane group 0/1
  const int ln   = lane & 15;

  // LDS: rg needs cols [ox-29, ox+50] (shift range +/-24 with col-group overreach), rows [oy-5, oy+22]
  __shared__ float sRG[28][80];
  __shared__ float sLG[28][32];
  __shared__ float sV[4][3][16][28];      // per-wave vertical-pass bounce buffer
  __shared__ float sLM[16][16];
  __shared__ float sLS[16][16];
  __shared__ float sBC[16][16];
  __shared__ float sBD[16][16];

  const float* lgb = lg + (size_t)b * H2 * W2;
  const float* rgb = rg + (size_t)b * H2 * W2;
  __builtin_prefetch(pred + ((size_t)b * H + oy * 2) * W + ox * 2, 0, 3);

  // cooperative zero-padded tile loads
  for (int idx = tid; idx < 28 * 80; idx += 128) {
    const int r = idx / 80, c = idx % 80;
    const int gy = oy - 5 + r, gx = ox - 29 + c;
    float v = 0.f;
    if (r < 26 && gy >= 0 && gy < H2 && gx >= 0 && gx < W2) v = rgb[gy * W2 + gx];
    sRG[r][c] = v;
  }
  for (int idx = tid; idx < 28 * 32; idx += 128) {
    const int r = idx / 32, c = idx % 32;
    const int gy = oy - 5 + r, gx = ox - 5 + c;
    float v = 0.f;
    if (r < 26 && gy >= 0 && gy < H2 && gx >= 0 && gx < W2) v = lgb[gy * W2 + gx];
    sLG[r][c] = v;
  }
  __syncthreads();

  // ---- shift-invariant lm / ls, computed once by wave 0 via WMMA box filters
  if (wv == 0) {
    v8f vA[2], vB[2];
#pragma unroll
    for (int g = 0; g < 2; ++g)
#pragma unroll
      for (int r = 0; r < 8; ++r) { vA[g][r] = 0.f; vB[g][r] = 0.f; }
#pragma unroll
    for (int g = 0; g < 2; ++g) {
#pragma unroll
      for (int j = 0; j < 7; ++j) {
        const int k0 = 4 * j + 2 * hi;
        v2f A;   // vertical band: BandV[m][k] = (m <= k <= m+10)
        A[0] = (k0     >= ln && k0     <= ln + 10) ? 1.f : 0.f;
        A[1] = (k0 + 1 >= ln && k0 + 1 <= ln + 10) ? 1.f : 0.f;
        const int col = ln + 16 * g;
        v2f Bl; Bl[0] = sLG[k0][col]; Bl[1] = sLG[k0 + 1][col];
        v2f B2; B2[0] = Bl[0] * Bl[0]; B2[1] = Bl[1] * Bl[1];
        vA[g] = wmma_f32(A, Bl, vA[g]);
        vB[g] = wmma_f32(A, B2, vB[g]);
      }
    }
#pragma unroll
    for (int g = 0; g < 2; ++g)
#pragma unroll
      for (int r = 0; r < 8; ++r) {
        const int m = r + 8 * hi, col = ln + 16 * g;
        if (col < 28) { sV[0][0][m][col] = vA[g][r]; sV[0][1][m][col] = vB[g][r]; }
      }
    v8f S0, S1;
#pragma unroll
    for (int r = 0; r < 8; ++r) { S0[r] = 0.f; S1[r] = 0.f; }
#pragma unroll
    for (int j = 0; j < 7; ++j) {
      const int k0 = 4 * j + 2 * hi;
      v2f Bb;  // horizontal band: BandH[k][n] = (n <= k <= n+10)
      Bb[0] = (k0     >= ln && k0     <= ln + 10) ? 1.f : 0.f;
      Bb[1] = (k0 + 1 >= ln && k0 + 1 <= ln + 10) ? 1.f : 0.f;
      v2f A0; A0[0] = sV[0][0][ln][k0]; A0[1] = sV[0][0][ln][k0 + 1];
      v2f A1; A1[0] = sV[0][1][ln][k0]; A1[1] = sV[0][1][ln][k0 + 1];
      S0 = wmma_f32(A0, Bb, S0);
      S1 = wmma_f32(A1, Bb, S1);
    }
#pragma unroll
    for (int r = 0; r < 8; ++r) {
      const int m = r + 8 * hi, n = ln;
      const float lm = S0[r] * (1.f / 121.f);
      const float e2 = S1[r] * (1.f / 121.f);
      sLM[m][n] = lm;
      sLS[m][n] = sqrtf(fmaxf(e2 - lm * lm, 1e-8f));
    }
  }
  __syncthreads();

  // ---- 48 shifts, 12 per wave
  float bc[8], bd[8];
#pragma unroll
  for (int r = 0; r < 8; ++r) { bc[r] = -1.f; bd[r] = 0.f; }

  for (int s = 0; s < 12; ++s) {
    const int i = wv + 4 * s;                       // 0..47
    const int d = (i < 24) ? (i - 24) : (i - 23);   // skip d==0
    // vertical pass: acc[g][f], f: 0=rs 1=rs^2 2=lg*rs
    v8f va[2][3];
#pragma unroll
    for (int g = 0; g < 2; ++g)
#pragma unroll
      for (int f = 0; f < 3; ++f)
#pragma unroll
        for (int r = 0; r < 8; ++r) va[g][f][r] = 0.f;
#pragma unroll
    for (int g = 0; g < 2; ++g) {
#pragma unroll
      for (int j = 0; j < 7; ++j) {
        const int k0 = 4 * j + 2 * hi;
        v2f A;
        A[0] = (k0     >= ln && k0     <= ln + 10) ? 1.f : 0.f;
        A[1] = (k0 + 1 >= ln && k0 + 1 <= ln + 10) ? 1.f : 0.f;
        const int col = ln + 16 * g;
        const int cc  = col + d + 24;               // shifted rg column in LDS
        v2f Br; Br[0] = sRG[k0][cc];          Br[1] = sRG[k0 + 1][cc];
        v2f Bq; Bq[0] = Br[0] * Br[0];        Bq[1] = Br[1] * Br[1];
        v2f Bc; Bc[0] = Br[0] * sLG[k0][col]; Bc[1] = Br[1] * sLG[k0 + 1][col];
        va[g][0] = wmma_f32(A, Br, va[g][0]);
        va[g][1] = wmma_f32(A, Bq, va[g][1]);
        va[g][2] = wmma_f32(A, Bc, va[g][2]);
      }
    }
#pragma unroll
    for (int g = 0; g < 2; ++g)
#pragma unroll
      for (int r = 0; r < 8; ++r) {
        const int m = r + 8 * hi, col = ln + 16 * g;
        if (col < 28) {
          sV[wv][0][m][col] = va[g][0][r];
          sV[wv][1][m][col] = va[g][1][r];
          sV[wv][2][m][col] = va[g][2][r];
        }
      }
    // horizontal pass
    v8f S[3];
#pragma unroll
    for (int f = 0; f < 3; ++f)
#pragma unroll
      for (int r = 0; r < 8; ++r) S[f][r] = 0.f;
#pragma unroll
    for (int f = 0; f < 3; ++f) {
#pragma unroll
      for (int j = 0; j < 7; ++j) {
        const int k0 = 4 * j + 2 * hi;
        v2f Bb;
        Bb[0] = (k0     >= ln && k0     <= ln + 10) ? 1.f : 0.f;
        Bb[1] = (k0 + 1 >= ln && k0 + 1 <= ln + 10) ? 1.f : 0.f;
        v2f Af; Af[0] = sV[wv][f][ln][k0]; Af[1] = sV[wv][f][ln][k0 + 1];
        S[f] = wmma_f32(Af, Bb, S[f]);
      }
    }
    // NCC + argmax update
#pragma unroll
    for (int r = 0; r < 8; ++r) {
      const int m = r + 8 * hi, n = ln;
      const float rm = S[0][r] * (1.f / 121.f);
      const float e2 = S[1][r] * (1.f / 121.f);
      const float cr = S[2][r] * (1.f / 121.f);
      const float rstd = sqrtf(fmaxf(e2 - rm * rm, 1e-8f));
      const float ncc = (cr - sLM[m][n] * rm) / (sLS[m][n] * rstd + 1e-8f);
      if (ncc > bc[r]) { bc[r] = ncc; bd[r] = (float)d; }
    }
  }

  // ---- merge the 4 waves' argmax in LDS
  for (int w = 0; w < 4; ++w) {
    if (wv == w) {
#pragma unroll
      for (int r = 0; r < 8; ++r) {
        const int m = r + 8 * hi, n = ln;
        if (w == 0) { sBC[m][n] = bc[r]; sBD[m][n] = bd[r]; }
        else if (bc[r] > sBC[m][n]) { sBC[m][n] = bc[r]; sBD[m][n] = bd[r]; }
      }
    }
    __syncthreads();
  }

  // ---- fused sign/magnitude loss at full res (nearest 2x upsample)
  if (wv == 0) {
    float s_sign = 0.f, s_mag = 0.f, s_cnt = 0.f;
#pragma unroll
    for (int r = 0; r < 8; ++r) {
      const int m = r + 8 * hi, n = ln;
      const float c  = sBC[m][n];
      const float nd = sBD[m][n] * 2.f;   // DS scale
      float nc = fmaxf(c, 0.f);
      nc = (nc > CONF_TH) ? nc : 0.f;
      if (nc > 0.f) {
        const float sgn = (nd > 0.f) ? 1.f : ((nd < 0.f) ? -1.f : 0.f);
        const int fy0 = (oy + m) * 2, fx0 = (ox + n) * 2;
#pragma unroll
        for (int dy = 0; dy < 2; ++dy)
#pragma unroll
          for (int dx = 0; dx < 2; ++dx) {
            const float p = pred[((size_t)b * H + fy0 + dy) * W + fx0 + dx];
            s_sign += fmaxf(-p * sgn, 0.f);
            s_mag  += nc * fabsf(p - nd);
            s_cnt  += 1.f;
          }
      }
    }
    for (int off = 16; off > 0; off >>= 1) {
      s_sign += __shfl_down(s_sign, off, 32);
      s_mag  += __shfl_down(s_mag,  off, 32);
      s_cnt  += __shfl_down(s_cnt,  off, 32);
    }
    if (lane == 0) {
      atomicAdd(&acc[6], s_sign);
      atomicAdd(&acc[7], s_mag);
      atomicAdd(&acc[8], s_cnt);
    }
  }
}

// ---------------------------------------------------------------- finalize scalar
__global__ void finalize_kernel(const float* acc, float* out) {
  if (threadIdx.x == 0 && blockIdx.x == 0) {
    const float gt_l    = acc[0] / fmaxf(acc[1], 1.f);
    const float photo_l = acc[2] / fmaxf(acc[3], 1.f);
    const float nn      = fmaxf(acc[8], 1.f);
    const float sm_l    = SIGN_W * (acc[6] / nn) + MAG_W * (acc[7] / nn);
    const float smooth_l = acc[4] / (float)(B * H * (W - 1))
                         + acc[5] / (float)(B * (H - 1) * W);
    out[0] = W_GT * gt_l + W_PHOTO * photo_l + W_SIGNMAG * sm_l + W_SMOOTH * smooth_l;
  }
}

// ---------------------------------------------------------------- launch
extern "C" void kernel_launch(void* const* d_in, const int* in_sizes, int n_in,
                              void* d_out, int out_size, void* d_ws, size_t ws_size,
                              hipStream_t stream) {
  const float* pred  = (const float*)d_in[0];
  const float* gt    = (const float*)d_in[1];
  const float* conf  = (const float*)d_in[2];
  const float* occ   = (const float*)d_in[3];
  const float* left  = (const float*)d_in[4];
  const float* right = (const float*)d_in[5];
  float* out = (float*)d_out;
  float* acc = (float*)d_ws;
  float* lg  = acc + 16;
  float* rg  = lg + (size_t)B * H2 * W2;

  init_kernel<<<1, 32, 0, stream>>>(acc);
  point_kernel<<<(B * H * W + 255) / 256, 256, 0, stream>>>(pred, gt, conf, occ, left, acc);
  photo_kernel<<<dim3(W / 16, H / 16, B), 256, 0, stream>>>(pred, left, right, acc);
  downsample_kernel<<<(B * H2 * W2 + 255) / 256, 256, 0, stream>>>(left, right, lg, rg);
  ncc_signmag_kernel<<<dim3(W2 / 16, H2 / 16, B), 128, 0, stream>>>(pred, lg, rg, acc);
  finalize_kernel<<<1, 1, 0, stream>>>(acc, out);
}